// Neural_Tensor_Layer_70394513981510
// MI455X (gfx1250) — compile-verified
//
#include <hip/hip_runtime.h>
#include <math.h>

typedef __attribute__((ext_vector_type(2))) float v2f;
typedef __attribute__((ext_vector_type(8))) float v8f;

#define B_ 64
#define S_ 64
#define H_ 512
#define F_ 32

// ---------------------------------------------------------------------------
// Stage 1: b_sum[b,h] = sum_s state_b[b,s,h]      (B*H threads)
// ---------------------------------------------------------------------------
__global__ void ntl_bsum_kernel(const float* __restrict__ sb,
                                float* __restrict__ bsum) {
    int idx = blockIdx.x * blockDim.x + threadIdx.x;   // 0 .. B*H-1
    int b = idx >> 9;          // / H
    int h = idx & (H_ - 1);
    const float* p = sb + (size_t)b * S_ * H_ + h;
    float acc = 0.f;
#pragma unroll 8
    for (int s = 0; s < S_; ++s) acc += p[(size_t)s * H_];
    bsum[idx] = acc;
}

// ---------------------------------------------------------------------------
// Stage 2: W'[b,k,f] = sum_h M[f,k,h] * bsum[b,h]  +  V[k,f]
// GEMM view: rows r=(f*H+k) of M_flat[16384,512]  x  bsum^T[512,64]
// One wave per 16-row strip; computes all 4 batch n-tiles (M read once).
// 1024 waves total.
// ---------------------------------------------------------------------------
__global__ void ntl_stage2_kernel(const float* __restrict__ Mt,
                                  const float* __restrict__ V,
                                  const float* __restrict__ bsum,
                                  float* __restrict__ Wp) {
    int wave = (blockIdx.x * blockDim.x + threadIdx.x) >> 5;  // 0..1023
    int lane = threadIdx.x & 31;
    int rowBase = wave * 16;                 // row tile in [0,16384)
    int m     = lane & 15;                   // A: row within tile
    int ncol  = lane & 15;                   // B: column within n-tile
    int khalf = (lane >> 4) * 2;             // K pair select (0 or 2)

    const float* arow = Mt + (size_t)(rowBase + m) * H_;
    const float* b0p  = bsum + (size_t)(0 * 16 + ncol) * H_;
    const float* b1p  = bsum + (size_t)(1 * 16 + ncol) * H_;
    const float* b2p  = bsum + (size_t)(2 * 16 + ncol) * H_;
    const float* b3p  = bsum + (size_t)(3 * 16 + ncol) * H_;

    v8f c0 = {}; v8f c1 = {}; v8f c2 = {}; v8f c3 = {};
#pragma unroll 4
    for (int k = 0; k < H_; k += 4) {
        v2f a  = *(const v2f*)(arow + k + khalf);
        v2f b0 = *(const v2f*)(b0p + k + khalf);
        v2f b1 = *(const v2f*)(b1p + k + khalf);
        v2f b2 = *(const v2f*)(b2p + k + khalf);
        v2f b3 = *(const v2f*)(b3p + k + khalf);
        c0 = __builtin_amdgcn_wmma_f32_16x16x4_f32(false, a, false, b0,
                                                   (short)0, c0, false, false);
        c1 = __builtin_amdgcn_wmma_f32_16x16x4_f32(false, a, false, b1,
                                                   (short)0, c1, false, false);
        c2 = __builtin_amdgcn_wmma_f32_16x16x4_f32(false, a, false, b2,
                                                   (short)0, c2, false, false);
        c3 = __builtin_amdgcn_wmma_f32_16x16x4_f32(false, a, false, b3,
                                                   (short)0, c3, false, false);
    }

    // Store: row r -> (f,k); col -> batch b.  W'[b,k,f] += V[k,f]
#pragma unroll
    for (int r = 0; r < 8; ++r) {
        int row = rowBase + r + (lane >> 4) * 8;
        int f = row >> 9;          // / H
        int k = row & (H_ - 1);
        float vadd = V[(size_t)k * F_ + f];
        size_t kf = (size_t)k * F_ + f;
        Wp[(size_t)(0 * 16 + ncol) * (H_ * F_) + kf] = c0[r] + vadd;
        Wp[(size_t)(1 * 16 + ncol) * (H_ * F_) + kf] = c1[r] + vadd;
        Wp[(size_t)(2 * 16 + ncol) * (H_ * F_) + kf] = c2[r] + vadd;
        Wp[(size_t)(3 * 16 + ncol) * (H_ * F_) + kf] = c3[r] + vadd;
    }
}

// ---------------------------------------------------------------------------
// Stage 3: out[b] = sigmoid(tanh( state_a[b] @ W'[b] + state_b[b] @ V_bot + bias ))
// Per batch: [S=64,H=512] x [H=512,F=32] -> 4x2 = 8 tiles; 512 waves total.
// ---------------------------------------------------------------------------
__global__ void ntl_stage3_kernel(const float* __restrict__ sa,
                                  const float* __restrict__ sb,
                                  const float* __restrict__ V,
                                  const float* __restrict__ bias,
                                  const float* __restrict__ Wp,
                                  float* __restrict__ out) {
    int gid  = (blockIdx.x * blockDim.x + threadIdx.x) >> 5;  // 0..511
    int lane = threadIdx.x & 31;
    int b     = gid >> 3;
    int t     = gid & 7;
    int tileS = t >> 1;        // 0..3
    int tileF = t & 1;         // 0..1
    int m     = lane & 15;
    int khalf = (lane >> 4) * 2;
    int s_row = tileS * 16 + m;
    int f_col = tileF * 16 + (lane & 15);

    const float* arow = sa + ((size_t)b * S_ + s_row) * H_;
    const float* brow = sb + ((size_t)b * S_ + s_row) * H_;
    const float* wb   = Wp + (size_t)b * H_ * F_;
    const float* vbot = V + (size_t)H_ * F_;

    v8f c = {};
#pragma unroll 4
    for (int k = 0; k < H_; k += 4) {
        v2f a = *(const v2f*)(arow + k + khalf);
        v2f bf;
        bf.x = wb[(size_t)(k + khalf + 0) * F_ + f_col];
        bf.y = wb[(size_t)(k + khalf + 1) * F_ + f_col];
        c = __builtin_amdgcn_wmma_f32_16x16x4_f32(false, a, false, bf,
                                                  (short)0, c, false, false);
    }
#pragma unroll 4
    for (int k = 0; k < H_; k += 4) {
        v2f a = *(const v2f*)(brow + k + khalf);
        v2f bf;
        bf.x = vbot[(size_t)(k + khalf + 0) * F_ + f_col];
        bf.y = vbot[(size_t)(k + khalf + 1) * F_ + f_col];
        c = __builtin_amdgcn_wmma_f32_16x16x4_f32(false, a, false, bf,
                                                  (short)0, c, false, false);
    }

#pragma unroll
    for (int r = 0; r < 8; ++r) {
        int s = tileS * 16 + r + (lane >> 4) * 8;
        size_t o = ((size_t)b * S_ + s) * F_ + f_col;
        float x  = c[r] + bias[o];
        float th = tanhf(x);
        out[o] = 1.0f / (1.0f + __expf(-th));
    }
}

// ---------------------------------------------------------------------------
extern "C" void kernel_launch(void* const* d_in, const int* in_sizes, int n_in,
                              void* d_out, int out_size, void* d_ws, size_t ws_size,
                              hipStream_t stream) {
    const float* state_a = (const float*)d_in[0];   // [B,S,H]
    const float* state_b = (const float*)d_in[1];   // [B,S,H]
    const float* M       = (const float*)d_in[2];   // [F,H,H]
    const float* V       = (const float*)d_in[3];   // [2H,F]
    const float* bias    = (const float*)d_in[4];   // [B,S,F]
    float* out = (float*)d_out;                     // [B,S,F]

    float* bsum = (float*)d_ws;                       // B*H floats = 128 KB
    float* Wp   = bsum + (size_t)B_ * H_;             // B*H*F floats = 4 MB

    // Stage 1: B*H = 32768 threads
    ntl_bsum_kernel<<<(B_ * H_) / 256, 256, 0, stream>>>(state_b, bsum);

    // Stage 2: 1024 waves = 32768 threads
    ntl_stage2_kernel<<<(1024 * 32) / 256, 256, 0, stream>>>(M, V, bsum, Wp);

    // Stage 3: 512 waves = 16384 threads
    ntl_stage3_kernel<<<(512 * 32) / 256, 256, 0, stream>>>(state_a, state_b, V,
                                                            bias, Wp, out);
}